// LuongAttention_24455543783460
// MI455X (gfx1250) — compile-verified
//
#include <hip/hip_runtime.h>
#include <math.h>

#define Bdim 16
#define Sdim 8192
#define Hdim 1024
#define NCHUNK 64
#define ROWS (Sdim / NCHUNK)   // 128 rows per context chunk
#define INV_SCALE 0.03125f     // 1 / sqrt(1024)

typedef __attribute__((ext_vector_type(2))) float v2f;
typedef __attribute__((ext_vector_type(4))) float f4;
typedef __attribute__((ext_vector_type(8))) float v8f;

// ---------------------------------------------------------------------------
// K1: tq[16,1024] = query[16,1024] @ W^T   via V_WMMA_F32_16X16X4_F32
// One wave per 16x16 output tile (M=16 is the whole batch), 64 tiles.
// A (16x4 f32): lane L holds row m=L%16, K pair {2*(L/16), 2*(L/16)+1}.
// B (4x16 f32): lane L holds col n=L%16, same K pair (B[k][n] = W[n][k],
// so a row of W gives a column of B -> contiguous float2 along K).
// ---------------------------------------------------------------------------
__global__ __launch_bounds__(32) void tq_wmma_kernel(
    const float* __restrict__ query, const float* __restrict__ W,
    float* __restrict__ tq) {
  const int ntile = blockIdx.x;        // 0..63
  const int lane  = threadIdx.x;       // 0..31
  const int lo = lane & 15;
  const int hi = lane >> 4;

  const float* aRow = query + (size_t)lo * Hdim;                 // m = lo
  const float* bRow = W + (size_t)(ntile * 16 + lo) * Hdim;      // n = lo

  v8f c = {};
  for (int k0 = 0; k0 < Hdim; k0 += 4) {
    const int k = k0 + 2 * hi;
    v2f a; a.x = aRow[k]; a.y = aRow[k + 1];
    v2f b; b.x = bRow[k]; b.y = bRow[k + 1];
    c = __builtin_amdgcn_wmma_f32_16x16x4_f32(
        /*neg_a=*/false, a, /*neg_b=*/false, b,
        /*c_mod=*/(short)0, c, /*reuse_a=*/false, /*reuse_b=*/false);
  }
  // C/D layout: VGPR r -> M = r + 8*hi, N = lo
#pragma unroll
  for (int r = 0; r < 8; ++r) {
    tq[(size_t)(r + 8 * hi) * Hdim + ntile * 16 + lo] = c[r];
  }
}

// ---------------------------------------------------------------------------
// K2: scores[b,s] = dot(tq[b], keys[b,s]) / 32, masked.  Streams 512 MB of
// keys with non-temporal b128 loads; tq row staged in LDS.
// ---------------------------------------------------------------------------
__global__ __launch_bounds__(256) void scores_kernel(
    const float* __restrict__ tq, const float* __restrict__ keys,
    const int* __restrict__ mask, float* __restrict__ scores) {
  __shared__ f4 tq_lds[Hdim / 4];
  const int chunksPerB = Sdim / 64;
  const int b = blockIdx.x / chunksPerB;
  const int chunk = blockIdx.x % chunksPerB;

  const f4* tq4 = (const f4*)(tq + (size_t)b * Hdim);
  for (int i = threadIdx.x; i < Hdim / 4; i += 256) tq_lds[i] = tq4[i];
  __syncthreads();

  const int wave = threadIdx.x >> 5;
  const int lane = threadIdx.x & 31;

  for (int r = wave; r < 64; r += 8) {
    const int s = chunk * 64 + r;
    const f4* kp = (const f4*)(keys + ((size_t)b * Sdim + s) * Hdim);
    float acc = 0.f;
#pragma unroll
    for (int i = 0; i < 8; ++i) {
      f4 kv = __builtin_nontemporal_load(&kp[lane + 32 * i]);
      f4 tv = tq_lds[lane + 32 * i];
      acc += kv.x * tv.x + kv.y * tv.y + kv.z * tv.z + kv.w * tv.w;
    }
#pragma unroll
    for (int off = 16; off; off >>= 1) acc += __shfl_xor(acc, off, 32);
    if (lane == 0) {
      float v = acc * INV_SCALE;
      if (mask[(size_t)b * Sdim + s] == 0) v = -1e9f;
      scores[(size_t)b * Sdim + s] = v;
    }
  }
}

// ---------------------------------------------------------------------------
// K3: per-batch softmax over S (scores are hot in L2). attn -> d_out.
// ---------------------------------------------------------------------------
__global__ __launch_bounds__(256) void softmax_kernel(
    const float* __restrict__ scores, float* __restrict__ attn) {
  const int b = blockIdx.x;
  __shared__ float red[8];
  const float* sc = scores + (size_t)b * Sdim;

  float m = -INFINITY;
  for (int i = threadIdx.x; i < Sdim; i += 256) m = fmaxf(m, sc[i]);
#pragma unroll
  for (int off = 16; off; off >>= 1) m = fmaxf(m, __shfl_xor(m, off, 32));
  if ((threadIdx.x & 31) == 0) red[threadIdx.x >> 5] = m;
  __syncthreads();
  m = red[0];
#pragma unroll
  for (int w = 1; w < 8; ++w) m = fmaxf(m, red[w]);
  __syncthreads();

  float sum = 0.f;
  for (int i = threadIdx.x; i < Sdim; i += 256) sum += __expf(sc[i] - m);
#pragma unroll
  for (int off = 16; off; off >>= 1) sum += __shfl_xor(sum, off, 32);
  if ((threadIdx.x & 31) == 0) red[threadIdx.x >> 5] = sum;
  __syncthreads();
  sum = 0.f;
#pragma unroll
  for (int w = 0; w < 8; ++w) sum += red[w];
  const float inv = 1.0f / sum;

  float* ap = attn + (size_t)b * Sdim;
  for (int i = threadIdx.x; i < Sdim; i += 256) ap[i] = __expf(sc[i] - m) * inv;
}

// ---------------------------------------------------------------------------
// K4: partial context. Each block: one (b, s-chunk of 128 rows); thread owns
// one float4 column of H. Streams 512 MB of values non-temporally.
// Deterministic: fixed partial slots, no atomics.
// ---------------------------------------------------------------------------
__global__ __launch_bounds__(256) void context_partial_kernel(
    const float* __restrict__ attn, const float* __restrict__ values,
    float* __restrict__ partial) {
  const int b = blockIdx.x / NCHUNK;
  const int chunk = blockIdx.x % NCHUNK;

  __shared__ float a_lds[ROWS];
  for (int i = threadIdx.x; i < ROWS; i += 256)
    a_lds[i] = attn[(size_t)b * Sdim + chunk * ROWS + i];
  __syncthreads();

  const int col = threadIdx.x;  // float4 column, 0..255
  const f4* vp = (const f4*)(values + ((size_t)b * Sdim + (size_t)chunk * ROWS) * Hdim);
  f4 acc = {0.f, 0.f, 0.f, 0.f};
  for (int r = 0; r < ROWS; ++r) {
    f4 v = __builtin_nontemporal_load(&vp[(size_t)r * (Hdim / 4) + col]);
    acc += v * a_lds[r];
  }
  f4* pp = (f4*)(partial + (size_t)(b * NCHUNK + chunk) * Hdim);
  pp[col] = acc;
}

// ---------------------------------------------------------------------------
// K5: reduce the 64 partials per (b,h). 16K outputs, trivial.
// ---------------------------------------------------------------------------
__global__ __launch_bounds__(256) void context_reduce_kernel(
    const float* __restrict__ partial, float* __restrict__ ctx) {
  const int idx = blockIdx.x * 256 + threadIdx.x;  // 0..B*H-1
  const int b = idx / Hdim;
  const int h = idx % Hdim;
  float sum = 0.f;
#pragma unroll 4
  for (int c = 0; c < NCHUNK; ++c)
    sum += partial[(size_t)(b * NCHUNK + c) * Hdim + h];
  ctx[idx] = sum;
}

extern "C" void kernel_launch(void* const* d_in, const int* in_sizes, int n_in,
                              void* d_out, int out_size, void* d_ws, size_t ws_size,
                              hipStream_t stream) {
  const float* query  = (const float*)d_in[0];   // [B,H]
  const float* keys   = (const float*)d_in[1];   // [B,S,H]
  const float* values = (const float*)d_in[2];   // [B,S,H]
  const int*   mask   = (const int*)  d_in[3];   // [B,S]
  const float* W      = (const float*)d_in[4];   // [H,H]

  float* out  = (float*)d_out;
  float* ctx  = out;                    // [B,H]  (first return)
  float* attn = out + Bdim * Hdim;      // [B,S]  (second return)

  // Workspace: tq (64 KB) | scores (512 KB) | partials (4 MB)
  float* tq      = (float*)d_ws;
  float* scores  = tq + Bdim * Hdim;
  float* partial = scores + Bdim * Sdim;

  tq_wmma_kernel<<<Hdim / 16, 32, 0, stream>>>(query, W, tq);
  scores_kernel<<<Bdim * (Sdim / 64), 256, 0, stream>>>(tq, keys, mask, scores);
  softmax_kernel<<<Bdim, 256, 0, stream>>>(scores, attn);
  context_partial_kernel<<<Bdim * NCHUNK, 256, 0, stream>>>(attn, values, partial);
  context_reduce_kernel<<<(Bdim * Hdim) / 256, 256, 0, stream>>>(partial, ctx);
}